// GraphSAGE_53472342835254
// MI455X (gfx1250) — compile-verified
//
#include <hip/hip_runtime.h>
#include <stdint.h>

#define DIM 128
#define NNODES 10000
#define NEDGES 640000
#define MTILES 625   // 10000 / 16 exactly
#define LROW 132     // LDS row stride in floats: 528B, 16B-aligned, bank-conflict-free

typedef __attribute__((ext_vector_type(2))) float v2f;
typedef __attribute__((ext_vector_type(8))) float v8f;

// ---------------------------------------------------------------------------
// Degree count: one fp32 atomic add per edge into cnt[dst].
// ---------------------------------------------------------------------------
__global__ void degree_kernel(const long long* __restrict__ dst,
                              float* __restrict__ cnt, int nedges) {
    int e = blockIdx.x * blockDim.x + threadIdx.x;
    if (e < nedges) {
        __hip_atomic_fetch_add(&cnt[(int)dst[e]], 1.0f,
                               __ATOMIC_RELAXED, __HIP_MEMORY_SCOPE_AGENT);
    }
}

// cnt -> 1/max(cnt,1) in place
__global__ void inv_kernel(float* __restrict__ c, int n) {
    int i = blockIdx.x * blockDim.x + threadIdx.x;
    if (i < n) c[i] = 1.0f / fmaxf(c[i], 1.0f);
}

// ---------------------------------------------------------------------------
// Scatter-add: one wave per edge, lane covers 4 floats (float4 gather + 4
// relaxed agent-scope fp32 atomics). Node table is L2-resident (5 MB).
// ---------------------------------------------------------------------------
__global__ void scatter_kernel(const float* __restrict__ x,
                               const long long* __restrict__ src,
                               const long long* __restrict__ dst,
                               float* __restrict__ aggr, int nedges) {
    int t = blockIdx.x * blockDim.x + threadIdx.x;
    int e = t >> 5;
    int lane = t & 31;
    if (e >= nedges) return;
    int s = (int)src[e];
    int d = (int)dst[e];
    float4 v = *(const float4*)(x + (size_t)s * DIM + lane * 4);
    float* o = aggr + (size_t)d * DIM + lane * 4;
    __hip_atomic_fetch_add(o + 0, v.x, __ATOMIC_RELAXED, __HIP_MEMORY_SCOPE_AGENT);
    __hip_atomic_fetch_add(o + 1, v.y, __ATOMIC_RELAXED, __HIP_MEMORY_SCOPE_AGENT);
    __hip_atomic_fetch_add(o + 2, v.z, __ATOMIC_RELAXED, __HIP_MEMORY_SCOPE_AGENT);
    __hip_atomic_fetch_add(o + 3, v.w, __ATOMIC_RELAXED, __HIP_MEMORY_SCOPE_AGENT);
}

// ---------------------------------------------------------------------------
// Fused SAGE layer: out = relu( (aggr * inv) @ Wl  +  x @ Wr  + b )
// 256 threads = 8 wave32; wave w owns output N-tile w (D = 8 * 16).
// Each wave register-caches all 32 K-fragments of Wl and Wr (128 VGPRs).
// A tiles (raw aggr + x, 16x128 each) are DMA'd global->LDS with
// GLOBAL_LOAD_ASYNC_TO_LDS_B128 (ASYNCcnt path), per-row 1/deg scale is
// applied at A-fragment read time.  64 x v_wmma_f32_16x16x4_f32 per M-tile.
// ---------------------------------------------------------------------------
__launch_bounds__(256)
__global__ void sage_gemm_kernel(const float* __restrict__ aggr,
                                 const float* __restrict__ inv,
                                 const float* __restrict__ x,
                                 const float* __restrict__ Wl,
                                 const float* __restrict__ bias,
                                 const float* __restrict__ Wr,
                                 float* __restrict__ out) {
    __shared__ float ldsA1[16 * LROW];   // aggregate tile (raw), padded rows
    __shared__ float ldsA2[16 * LROW];   // self-feature tile, padded rows

    const int tid  = threadIdx.x;
    const int lane = tid & 31;
    const int wave = tid >> 5;                  // = N-tile index 0..7
    const int nsub = lane & 15;                 // N within tile / M-row for A frag
    const int khalf = (lane >> 4) * 2;          // K offset 0 or 2 within 4-step
    const int ncol = wave * 16 + nsub;          // global output column

    // --- register-cache W fragments (B-operand layout: N striped over lanes,
    //     VGPR0 = K rows {0,2}, VGPR1 = K rows {1,3} per 4-wide K step) ---
    v2f wl[32], wr[32];
#pragma unroll
    for (int ks = 0; ks < 32; ++ks) {
        int k0 = ks * 4 + khalf;
        wl[ks].x = Wl[(size_t)k0 * DIM + ncol];
        wl[ks].y = Wl[(size_t)(k0 + 1) * DIM + ncol];
        wr[ks].x = Wr[(size_t)k0 * DIM + ncol];
        wr[ks].y = Wr[(size_t)(k0 + 1) * DIM + ncol];
    }
    const float bn = bias[ncol];

    // low 32 bits of a flat shared pointer == LDS byte offset (aperture is in
    // the high half), usable directly as the async-copy LDS destination.
    const unsigned l1base = (unsigned)(uintptr_t)&ldsA1[0];
    const unsigned l2base = (unsigned)(uintptr_t)&ldsA2[0];

    for (int mt = blockIdx.x; mt < MTILES; mt += gridDim.x) {
        const int row0 = mt * 16;

        __syncthreads();   // previous iteration's LDS reads must drain

        // ---- async DMA staging: 2 x (16 rows x 512B) global -> LDS --------
        // 512 x 16B chunks per tile, 256 threads -> 2 chunks each.
#pragma unroll
        for (int j = 0; j < 2; ++j) {
            int i  = tid + j * 256;             // 0..511
            int r  = i >> 5;                    // row 0..15
            int c4 = i & 31;                    // 16B chunk within row
            unsigned voff = (unsigned)((row0 + r) * (DIM * 4) + c4 * 16);
            unsigned loff = (unsigned)(r * (LROW * 4) + c4 * 16);
            unsigned d1 = l1base + loff;
            unsigned d2 = l2base + loff;
            asm volatile("global_load_async_to_lds_b128 %0, %1, %2"
                         :: "v"(d1), "v"(voff), "s"(aggr) : "memory");
            asm volatile("global_load_async_to_lds_b128 %0, %1, %2"
                         :: "v"(d2), "v"(voff), "s"(x) : "memory");
        }
        asm volatile("s_wait_asynccnt 0" ::: "memory");
        __syncthreads();

        const float sinv = inv[row0 + nsub];    // per-row mean scale (A rows)

        v8f acc = {};
#pragma unroll
        for (int ks = 0; ks < 32; ++ks) {
            int k0 = ks * 4 + khalf;
            v2f a1, a2;
            a1.x = ldsA1[nsub * LROW + k0] * sinv;
            a1.y = ldsA1[nsub * LROW + k0 + 1] * sinv;
            a2.x = ldsA2[nsub * LROW + k0];
            a2.y = ldsA2[nsub * LROW + k0 + 1];
            acc = __builtin_amdgcn_wmma_f32_16x16x4_f32(
                false, a1, false, wl[ks], (short)0, acc, false, false);
            acc = __builtin_amdgcn_wmma_f32_16x16x4_f32(
                false, a2, false, wr[ks], (short)0, acc, false, false);
        }

        // C/D layout: VGPR v -> M = v + (lane>=16 ? 8 : 0); N = ncol
        const int mbase = row0 + ((lane >> 4) << 3);
#pragma unroll
        for (int v = 0; v < 8; ++v) {
            float r = acc[v] + bn;
            out[(size_t)(mbase + v) * DIM + ncol] = fmaxf(r, 0.0f);
        }
    }
}

// ---------------------------------------------------------------------------
extern "C" void kernel_launch(void* const* d_in, const int* in_sizes, int n_in,
                              void* d_out, int out_size, void* d_ws, size_t ws_size,
                              hipStream_t stream) {
    (void)in_sizes; (void)n_in; (void)out_size; (void)ws_size;

    const float*     node = (const float*)d_in[0];
    const long long* ei   = (const long long*)d_in[1];  // int64 edge_index [2,E]
    const float*     W1l  = (const float*)d_in[2];
    const float*     b1   = (const float*)d_in[3];
    const float*     W1r  = (const float*)d_in[4];
    const float*     W2l  = (const float*)d_in[5];
    const float*     b2   = (const float*)d_in[6];
    const float*     W2r  = (const float*)d_in[7];
    float* out = (float*)d_out;

    float* ws   = (float*)d_ws;
    float* inv  = ws;                             // 10000 floats (cnt -> 1/cnt)
    float* aggr = ws + 16384;                     // 10000*128 floats
    float* h1   = aggr + (size_t)NNODES * DIM;    // 10000*128 floats

    const long long* srcp = ei;
    const long long* dstp = ei + NEDGES;

    const size_t aggr_bytes = (size_t)NNODES * DIM * sizeof(float);

    // shared degree counts (edges identical for both layers)
    hipMemsetAsync(inv, 0, NNODES * sizeof(float), stream);
    degree_kernel<<<(NEDGES + 255) / 256, 256, 0, stream>>>(dstp, inv, NEDGES);
    inv_kernel<<<(NNODES + 255) / 256, 256, 0, stream>>>(inv, NNODES);

    // ---- layer 1 ----
    hipMemsetAsync(aggr, 0, aggr_bytes, stream);
    scatter_kernel<<<(NEDGES * 32) / 256, 256, 0, stream>>>(node, srcp, dstp, aggr, NEDGES);
    sage_gemm_kernel<<<160, 256, 0, stream>>>(aggr, inv, node, W1l, b1, W1r, h1);

    // ---- layer 2 ----
    hipMemsetAsync(aggr, 0, aggr_bytes, stream);
    scatter_kernel<<<(NEDGES * 32) / 256, 256, 0, stream>>>(h1, srcp, dstp, aggr, NEDGES);
    sage_gemm_kernel<<<160, 256, 0, stream>>>(aggr, inv, h1, W2l, b2, W2r, out);
}